// charge_5265629904896
// MI455X (gfx1250) — compile-verified
//
#include <hip/hip_runtime.h>

typedef __attribute__((ext_vector_type(2))) float v2f;
typedef __attribute__((ext_vector_type(8))) float v8f;

#define N_PTS 131072
#define DIM 64
#define NQ 16
#define TILE 16
#define KSTEPS (DIM / 4) // 16 WMMA_F32_16X16X4_F32 steps cover K=64

// Monotone float -> uint mapping (ascending order preserved)
__device__ __forceinline__ unsigned ford(float f) {
  unsigned u = __float_as_uint(f);
  return (u & 0x80000000u) ? ~u : (u | 0x80000000u);
}

__global__ void nn_init(unsigned long long* best) {
  if (threadIdx.x < NQ) best[threadIdx.x] = ~0ull;
}

__global__ __launch_bounds__(256) void nn_argmin(
    const float* __restrict__ u, const float* __restrict__ loc,
    unsigned long long* __restrict__ best) {
  const unsigned lane = threadIdx.x & 31u;
  const unsigned waveInBlock = threadIdx.x >> 5;
  const unsigned wavesPerBlock = blockDim.x >> 5;
  const unsigned waveId = blockIdx.x * wavesPerBlock + waveInBlock;
  const unsigned numWaves = gridDim.x * wavesPerBlock;

  const unsigned row = lane & 15u; // M for A, N for B/C/D
  const unsigned hi = lane >> 4;   // half-wave: selects K pair {0,1} vs {2,3}

  // A-matrix (queries) in ISA 16x4 f32 layout, preloaded for all 16 k-steps:
  // lane holds U[row][4k + 2*hi + {0,1}] in {VGPR0, VGPR1}
  v2f A[KSTEPS];
  for (int k = 0; k < KSTEPS; ++k) {
    const float* p = u + row * DIM + 4 * k + 2 * hi;
    A[k].x = p[0];
    A[k].y = p[1];
  }

  float bestScore[8];
  unsigned bestIdx[8];
  for (int r = 0; r < 8; ++r) {
    bestScore[r] = __builtin_inff();
    bestIdx[r] = 0u;
  }

  const unsigned numTiles = N_PTS / TILE; // 8192, divides evenly
  for (unsigned t = waveId; t < numTiles; t += numWaves) { // wave-uniform
    const unsigned base = t * TILE;
    v8f acc = {};
    float sumsq = 0.f;
    // B-matrix 4x16 f32 layout per k-step: lane n=row supplies
    // Loc[base+n][4k + 2*hi + {0,1}] in {VGPR0, VGPR1}
    const float* rp = loc + (size_t)(base + row) * DIM + 2 * hi;
    for (int k = 0; k < KSTEPS; ++k) {
      v2f b;
      b.x = rp[4 * k];
      b.y = rp[4 * k + 1];
      sumsq = __builtin_fmaf(b.x, b.x, sumsq);
      sumsq = __builtin_fmaf(b.y, b.y, sumsq);
      // D = A x B + C  (16x16 f32 accumulator, 8 VGPRs)
      acc = __builtin_amdgcn_wmma_f32_16x16x4_f32(
          /*neg_a=*/false, A[k], /*neg_b=*/false, b,
          /*c_mod=*/(short)0, acc, /*reuse_a=*/false, /*reuse_b=*/false);
    }
    // complete ||loc_n||^2: each half-wave has half the K range
    sumsq += __shfl_xor(sumsq, 16, 32);

    // C/D layout: VGPR r -> M = r + 8*hi, lane -> N = lane & 15
    const unsigned n = base + row;
    for (int r = 0; r < 8; ++r) {
      float score = __builtin_fmaf(-2.0f, acc[r], sumsq); // ||l||^2 - 2 u.l
      if (score < bestScore[r]) {
        bestScore[r] = score;
        bestIdx[r] = n;
      }
    }
  }

  // Per-query reduction: lanes 0-15 hold query m=r, lanes 16-31 hold m=r+8.
  // xor by 1/2/4/8 stays within each 16-lane half.
  for (int r = 0; r < 8; ++r) {
    unsigned long long key =
        ((unsigned long long)ford(bestScore[r]) << 32) |
        (unsigned long long)bestIdx[r];
    for (int s = 1; s < 16; s <<= 1) {
      unsigned long long o = __shfl_xor(key, s, 32);
      if (o < key) key = o;
    }
    if (row == 0) {
      const unsigned m = (unsigned)r + 8u * hi;
      atomicMin(&best[m], key); // global_atomic_min_u64
    }
  }
}

__global__ void nn_gather(const unsigned long long* __restrict__ best,
                          const float* __restrict__ val,
                          float* __restrict__ out) {
  const unsigned tid = blockIdx.x * blockDim.x + threadIdx.x;
  if (tid < NQ * DIM) {
    const unsigned b = tid >> 6;
    const unsigned d = tid & 63u;
    const unsigned idx = (unsigned)(best[b] & 0xFFFFFFFFull);
    out[tid] = val[(size_t)idx * DIM + d];
  }
}

extern "C" void kernel_launch(void* const* d_in, const int* in_sizes, int n_in,
                              void* d_out, int out_size, void* d_ws,
                              size_t ws_size, hipStream_t stream) {
  const float* u = (const float*)d_in[0];   // [16, 64]
  const float* loc = (const float*)d_in[1]; // [131072, 64]
  const float* val = (const float*)d_in[2]; // [131072, 64]
  // d_in[3] is p (==2), L2 hardcoded.
  unsigned long long* best = (unsigned long long*)d_ws; // 16 packed slots
  float* out = (float*)d_out;

  nn_init<<<1, 64, 0, stream>>>(best);
  nn_argmin<<<256, 256, 0, stream>>>(u, loc, best);
  nn_gather<<<(NQ * DIM + 255) / 256, 256, 0, stream>>>(best, val, out);
}